// ZSVGP_17746804867832
// MI455X (gfx1250) — compile-verified
//
#include <hip/hip_runtime.h>
#include <math.h>

#define M_IND 1024
#define N_PTS 8192
#define A_GP  8
#define D_FEAT 16
#define JITTER_F 1e-4f
#define LOG2PI_F 1.8378770664093453f

typedef __attribute__((ext_vector_type(2))) float v2f;
typedef __attribute__((ext_vector_type(8))) float v8f;

// ---------------------------------------------------------------------------
// CDNA5 fp32 WMMA: D(16x16) = A(16x4) * B(4x16) + C, wave32.
// ---------------------------------------------------------------------------
__device__ __forceinline__ v8f wmma4(v2f a, v2f b, v8f c) {
  return __builtin_amdgcn_wmma_f32_16x16x4_f32(false, a, false, b, (short)0, c,
                                               false, false);
}

// C(Mr x Nc) = [Base +] A^T * [diag(W)] * B ; A,B are K x (Mr|Nc) row-major.
// One wave computes a 32x32 tile (2x2 WMMA tiles, 4 wmma per K-step).
// Batched via blockIdx.z with element strides sA/sB/sW/sC (Base shared).
template <bool HASW, bool HASBASE>
__global__ void wmma_gemm_tn(const float* __restrict__ A,
                             const float* __restrict__ B,
                             const float* __restrict__ W,
                             const float* __restrict__ Base,
                             float* __restrict__ C,
                             int K, int Mr, int Nc,
                             long long sA, long long sB, long long sW,
                             long long sC) {
  const long long bz = blockIdx.z;
  A += bz * sA;
  B += bz * sB;
  C += bz * sC;
  if (HASW) W += bz * sW;
  const int tm = blockIdx.y * 32, tn = blockIdx.x * 32;
  const int lane = threadIdx.x & 31, hl = lane >> 4, l = lane & 15;
  const int m0 = tm + l, m1 = m0 + 16;
  const int n0 = tn + l, n1 = n0 + 16;

  v8f c00, c01, c10, c11;
#pragma unroll
  for (int r = 0; r < 8; ++r) {
    const int row = r + hl * 8;
    if (HASBASE) {
      c00[r] = Base[(size_t)(tm + row) * Nc + n0];
      c01[r] = Base[(size_t)(tm + row) * Nc + n1];
      c10[r] = Base[(size_t)(tm + 16 + row) * Nc + n0];
      c11[r] = Base[(size_t)(tm + 16 + row) * Nc + n1];
    } else {
      c00[r] = 0.f; c01[r] = 0.f; c10[r] = 0.f; c11[r] = 0.f;
    }
  }

  // Pointer-bumped K loop: rows ka = k0 + hl*2 and ka+1.
  const float* Ap0 = A + (size_t)(hl * 2) * Mr;
  const float* Ap1 = Ap0 + Mr;
  const float* Bp0 = B + (size_t)(hl * 2) * Nc;
  const float* Bp1 = Bp0 + Nc;
  const float* Wp = HASW ? (W + hl * 2) : nullptr;

#pragma unroll 2
  for (int k0 = 0; k0 < K; k0 += 4) {
    v2f w;
    if (HASW) {
      w = *(const v2f*)Wp;          // W[ka], W[ka+1] (8B aligned: ka even)
      Wp += 4;
    } else {
      w.x = 1.f; w.y = 1.f;
    }
    v2f a0, a1, b0, b1;
    a0.x = w.x * Ap0[m0]; a0.y = w.y * Ap1[m0];
    a1.x = w.x * Ap0[m1]; a1.y = w.y * Ap1[m1];
    b0.x = Bp0[n0];       b0.y = Bp1[n0];
    b1.x = Bp0[n1];       b1.y = Bp1[n1];
    c00 = wmma4(a0, b0, c00);
    c01 = wmma4(a0, b1, c01);
    c10 = wmma4(a1, b0, c10);
    c11 = wmma4(a1, b1, c11);
    Ap0 += 4 * (size_t)Mr; Ap1 += 4 * (size_t)Mr;
    Bp0 += 4 * (size_t)Nc; Bp1 += 4 * (size_t)Nc;
  }

#pragma unroll
  for (int r = 0; r < 8; ++r) {
    const int row = r + hl * 8;
    C[(size_t)(tm + row) * Nc + n0] = c00[r];
    C[(size_t)(tm + row) * Nc + n1] = c01[r];
    C[(size_t)(tm + 16 + row) * Nc + n0] = c10[r];
    C[(size_t)(tm + 16 + row) * Nc + n1] = c11[r];
  }
}

// C(rows x Nc) = A(rows x K) * B(K x Nc), row-major; 32x32 tile per wave.
__global__ void wmma_gemm_nn(const float* __restrict__ A,
                             const float* __restrict__ B,
                             float* __restrict__ C, int K, int Nc) {
  const int tm = blockIdx.y * 32, tn = blockIdx.x * 32;
  const int lane = threadIdx.x & 31, hl = lane >> 4, l = lane & 15;
  const int n0 = tn + l, n1 = n0 + 16;

  v8f c00, c01, c10, c11;
#pragma unroll
  for (int r = 0; r < 8; ++r) { c00[r] = 0.f; c01[r] = 0.f; c10[r] = 0.f; c11[r] = 0.f; }

  const float* Ar0 = A + (size_t)(tm + l) * K + hl * 2;        // k-pair, 8B aligned
  const float* Ar1 = A + (size_t)(tm + 16 + l) * K + hl * 2;
  const float* Bp0 = B + (size_t)(hl * 2) * Nc;
  const float* Bp1 = Bp0 + Nc;

#pragma unroll 2
  for (int k0 = 0; k0 < K; k0 += 4) {
    v2f a0 = *(const v2f*)Ar0;
    v2f a1 = *(const v2f*)Ar1;
    v2f b0, b1;
    b0.x = Bp0[n0]; b0.y = Bp1[n0];
    b1.x = Bp0[n1]; b1.y = Bp1[n1];
    c00 = wmma4(a0, b0, c00);
    c01 = wmma4(a0, b1, c01);
    c10 = wmma4(a1, b0, c10);
    c11 = wmma4(a1, b1, c11);
    Ar0 += 4; Ar1 += 4;
    Bp0 += 4 * (size_t)Nc; Bp1 += 4 * (size_t)Nc;
  }

#pragma unroll
  for (int r = 0; r < 8; ++r) {
    const int row = r + hl * 8;
    C[(size_t)(tm + row) * Nc + n0] = c00[r];
    C[(size_t)(tm + row) * Nc + n1] = c01[r];
    C[(size_t)(tm + 16 + row) * Nc + n0] = c10[r];
    C[(size_t)(tm + 16 + row) * Nc + n1] = c11[r];
  }
}

// ---------------------------------------------------------------------------
// Elementwise / support kernels
// ---------------------------------------------------------------------------
__global__ void k_zero(float* p, int n) {
  int i = blockIdx.x * blockDim.x + threadIdx.x;
  if (i < n) p[i] = 0.f;
}

__global__ void k_cauchy_mm(const float* __restrict__ U,
                            const float* __restrict__ ks,
                            float* __restrict__ Kmm) {
  int idx = blockIdx.x * blockDim.x + threadIdx.x;
  if (idx >= M_IND * M_IND) return;
  int i = idx / M_IND, j = idx % M_IND;
  float dx = U[2 * i] - U[2 * j], dy = U[2 * i + 1] - U[2 * j + 1];
  float d2 = fmaxf(dx * dx + dy * dy, 0.f);
  float s = fmaxf(ks[0], 1e-10f);
  float v = 1.f / (1.f + d2 / s);
  if (i == j) v += JITTER_F;
  Kmm[idx] = v;
}

__global__ void k_cauchy_nm(const float* __restrict__ X,
                            const float* __restrict__ U,
                            const float* __restrict__ ks,
                            float* __restrict__ Knm) {
  int idx = blockIdx.x * blockDim.x + threadIdx.x;
  if (idx >= N_PTS * M_IND) return;
  int n = idx / M_IND, j = idx % M_IND;
  float dx = X[2 * n] - U[2 * j], dy = X[2 * n + 1] - U[2 * j + 1];
  float d2 = fmaxf(dx * dx + dy * dy, 0.f);
  float s = fmaxf(ks[0], 1e-10f);
  Knm[idx] = 1.f / (1.f + d2 / s);
}

// precT[a*N + n] = 1 / x_var[n*16 + a]
__global__ void k_precT(const float* __restrict__ x_var, float* __restrict__ pT) {
  int idx = blockIdx.x * blockDim.x + threadIdx.x;
  if (idx >= A_GP * N_PTS) return;
  int a = idx / N_PTS, n = idx % N_PTS;
  pT[idx] = 1.f / x_var[(size_t)n * D_FEAT + a];
}

// In-place lower Cholesky of an M x M SPD matrix; one workgroup per matrix.
__global__ void k_chol(float* __restrict__ Abase, float* __restrict__ logdets,
                       int ldetOff) {
  float* A = Abase + (size_t)blockIdx.x * M_IND * M_IND;
  const int t = threadIdx.x;
  __shared__ float s_d;
  __shared__ float s_col[M_IND];
  float ld = 0.f;
  for (int k = 0; k < M_IND; ++k) {
    if (t == 0) {
      float d = sqrtf(fmaxf(A[(size_t)k * M_IND + k], 1e-30f));
      A[(size_t)k * M_IND + k] = d;
      s_d = d;
      ld += 2.f * logf(d);
    }
    __syncthreads();
    const int i = k + 1 + t;
    float lik = 0.f;
    if (i < M_IND) {
      lik = A[(size_t)i * M_IND + k] / s_d;
      A[(size_t)i * M_IND + k] = lik;
    }
    s_col[t] = lik;
    __syncthreads();
    if (i < M_IND) {
      for (int j = k + 1; j <= i; ++j)
        A[(size_t)i * M_IND + j] -= lik * s_col[j - k - 1];
    }
    __syncthreads();
  }
  if (t == 0) logdets[ldetOff + blockIdx.x] = ld;
}

// Inverse of lower-triangular L (dense output, zeros above diagonal).
// One workgroup per matrix; thread j solves column j by forward substitution.
__global__ void k_trinv(const float* __restrict__ Lbase, float* __restrict__ Ob) {
  const float* L = Lbase + (size_t)blockIdx.x * M_IND * M_IND;
  float* O = Ob + (size_t)blockIdx.x * M_IND * M_IND;
  const int j = threadIdx.x;
  for (int i = 0; i < j; ++i) O[(size_t)i * M_IND + j] = 0.f;
  O[(size_t)j * M_IND + j] = 1.f / L[(size_t)j * M_IND + j];
  for (int i = j + 1; i < M_IND; ++i) {
    float s = 0.f;
    for (int k = j; k < i; ++k)
      s += L[(size_t)i * M_IND + k] * O[(size_t)k * M_IND + j];
    O[(size_t)i * M_IND + j] = -s / L[(size_t)i * M_IND + i];
  }
}

// r[a*M + m] = sum_n Knm[n,m] * x_loc[n,a] * precT[a,n]
__global__ void k_r(const float* __restrict__ Knm, const float* __restrict__ x_loc,
                    const float* __restrict__ pT, float* __restrict__ r) {
  int m = blockIdx.x * blockDim.x + threadIdx.x;
  int a = blockIdx.y;
  if (m >= M_IND) return;
  float s = 0.f;
  for (int n = 0; n < N_PTS; ++n)
    s += Knm[(size_t)n * M_IND + m] * x_loc[(size_t)n * D_FEAT + a] *
         pT[(size_t)a * N_PTS + n];
  r[a * M_IND + m] = s;
}

// out[a*M+m] = sum_k Mat[(a*sMat) + m*M + k] * vec[a*M + k]
__global__ void k_matvec(const float* __restrict__ Mat,
                         const float* __restrict__ vec,
                         float* __restrict__ out, long long sMat) {
  int m = blockIdx.x * blockDim.x + threadIdx.x;
  int a = blockIdx.y;
  if (m >= M_IND) return;
  const float* Ma = Mat + (long long)a * sMat + (size_t)m * M_IND;
  const float* v = vec + (size_t)a * M_IND;
  float s = 0.f;
  for (int k = 0; k < M_IND; ++k) s += Ma[k] * v[k];
  out[a * M_IND + m] = s;
}

// gp_p_m[n*8+a] = sum_m Knm[n,m] * si_r[a,m]
__global__ void k_gppm(const float* __restrict__ Knm,
                       const float* __restrict__ sir, float* __restrict__ pm) {
  int idx = blockIdx.x * blockDim.x + threadIdx.x;
  if (idx >= N_PTS * A_GP) return;
  int n = idx >> 3, a = idx & 7;
  const float* kr = Knm + (size_t)n * M_IND;
  const float* sr = sir + (size_t)a * M_IND;
  float s = 0.f;
  for (int m = 0; m < M_IND; ++m) s += kr[m] * sr[m];
  pm[idx] = s;
}

// out[n*oStride + oOff] = sum_k X[n*K+k] * Y[n*K+k]
__global__ void k_rowdot(const float* __restrict__ X, const float* __restrict__ Y,
                         float* __restrict__ out, int K, int oStride, int oOff) {
  int n = blockIdx.x * blockDim.x + threadIdx.x;
  if (n >= N_PTS) return;
  const float* x = X + (size_t)n * K;
  const float* y = Y + (size_t)n * K;
  float s = 0.f;
  for (int k = 0; k < K; ++k) s += x[k] * y[k];
  out[(size_t)n * oStride + oOff] = s;
}

// acc[accOff+a] += dot(X + a*sX, Y + a*sY) over `count` elements
__global__ void k_bdot(const float* __restrict__ X, const float* __restrict__ Y,
                       float* __restrict__ acc, int count, long long sX,
                       long long sY, int accOff) {
  int a = blockIdx.y;
  const float* x = X + (long long)a * sX;
  const float* y = Y + (long long)a * sY;
  __shared__ float red[256];
  float s = 0.f;
  for (int i = blockIdx.x * blockDim.x + threadIdx.x; i < count;
       i += gridDim.x * blockDim.x)
    s += x[i] * y[i];
  red[threadIdx.x] = s;
  __syncthreads();
  for (int o = 128; o > 0; o >>= 1) {
    if ((int)threadIdx.x < o) red[threadIdx.x] += red[threadIdx.x + o];
    __syncthreads();
  }
  if (threadIdx.x == 0) atomicAdd(&acc[accOff + a], red[0]);
}

// Write p_m / p_v, accumulate the 6 scalar sums.
// acc: 0=Ktilde 1=trace 2=quad 3=sum_log_gpvar 4=ce_sum 5=gauss_kl
__global__ void k_assemble(const float* __restrict__ x_loc,
                           const float* __restrict__ x_var,
                           const float* __restrict__ gppm,
                           const float* __restrict__ diag1,
                           const float* __restrict__ diag2,
                           float* __restrict__ out, float* __restrict__ acc) {
  int idx = blockIdx.x * blockDim.x + threadIdx.x;  // 0 .. N*16-1
  int n = idx >> 4, c = idx & 15;
  float mu = x_loc[idx], var = x_var[idx];
  float vals[6] = {0.f, 0.f, 0.f, 0.f, 0.f, 0.f};
  float pm, pv;
  if (c < A_GP) {
    float prec = 1.f / var;
    pm = gppm[n * A_GP + c];
    float d1 = diag1[n * A_GP + c];
    float d2 = diag2[n];
    pv = 1.f + d1 - d2;
    float lv = logf(var);
    vals[0] = prec * (1.f - d2);
    vals[1] = prec * d1;
    float e = mu - pm;
    vals[2] = prec * e * e;
    vals[3] = lv;
    vals[4] =
        -0.5f * (LOG2PI_F + lv + (pv + pm * pm - 2.f * pm * mu + mu * mu) * prec);
  } else {
    pm = mu;
    pv = var;
    vals[5] = 0.5f * (var + mu * mu - 1.f) - 0.5f * logf(var);
  }
  out[idx] = pm;
  out[N_PTS * D_FEAT + idx] = pv;
  __shared__ float red[256];
  for (int q = 0; q < 6; ++q) {
    red[threadIdx.x] = vals[q];
    __syncthreads();
    for (int o = 128; o > 0; o >>= 1) {
      if ((int)threadIdx.x < o) red[threadIdx.x] += red[threadIdx.x + o];
      __syncthreads();
    }
    if (threadIdx.x == 0) atomicAdd(&acc[q], red[0]);
    __syncthreads();
  }
}

// logdets[0]=logdet Kmm, logdets[1+a]=logdet sigma_l[a]
// acc[6+a]=tr(Sinv Kmm), acc[14+a]=si_r^T Kmm si_r
__global__ void k_finalize(const float* __restrict__ acc,
                           const float* __restrict__ logdets,
                           float* __restrict__ out) {
  if (threadIdx.x != 0 || blockIdx.x != 0) return;
  float recon = -0.5f * (acc[0] + acc[1] + acc[3] +
                         (float)A_GP * (float)N_PTS * LOG2PI_F + acc[2]);
  float klsum = 0.f;
  for (int a = 0; a < A_GP; ++a)
    klsum += 0.5f * (logdets[1 + a] - logdets[0] - (float)M_IND + acc[6 + a] +
                     acc[14 + a]);
  float inside = recon - klsum;  // N / N_TRAIN == 1
  out[2 * N_PTS * D_FEAT] = acc[4] - inside;  // x_gp_kl
  out[2 * N_PTS * D_FEAT + 1] = acc[5];       // x_gauss_kl
}

// ---------------------------------------------------------------------------
extern "C" void kernel_launch(void* const* d_in, const int* in_sizes, int n_in,
                              void* d_out, int out_size, void* d_ws,
                              size_t ws_size, hipStream_t stream) {
  (void)in_sizes; (void)n_in; (void)out_size; (void)ws_size;
  const float* x_loc = (const float*)d_in[0];
  const float* x_var = (const float*)d_in[1];
  const float* xpos  = (const float*)d_in[2];
  const float* ind   = (const float*)d_in[3];
  const float* ks    = (const float*)d_in[4];
  float* out = (float*)d_out;

  const size_t MM = (size_t)M_IND * M_IND;  // 1,048,576
  const size_t NM = (size_t)N_PTS * M_IND;  // 8,388,608
  float* ws      = (float*)d_ws;
  float* Kmm     = ws;
  float* KmmChol = Kmm + MM;
  float* KmmLinv = KmmChol + MM;
  float* KmmInv  = KmmLinv + MM;
  float* Knm     = KmmInv + MM;
  float* Sig     = Knm + NM;     // 8 matrices (chol'd in place)
  float* SigLinv = Sig + 8 * MM;
  float* SigInv  = SigLinv + 8 * MM;
  float* Q       = SigInv + 8 * MM;  // N x M scratch, reused
  float* precT   = Q + NM;
  float* rv      = precT + (size_t)A_GP * N_PTS;
  float* sir     = rv + A_GP * M_IND;
  float* KmmSir  = sir + A_GP * M_IND;
  float* gppm    = KmmSir + A_GP * M_IND;
  float* diag1   = gppm + (size_t)N_PTS * A_GP;
  float* diag2   = diag1 + (size_t)N_PTS * A_GP;
  float* logdets = diag2 + N_PTS;  // 16 slots
  float* acc     = logdets + 16;   // 32 slots

  dim3 b32(32), b256(256), b1024(1024);

  k_zero<<<1, 64, 0, stream>>>(logdets, 48);
  k_cauchy_mm<<<(M_IND * M_IND + 255) / 256, b256, 0, stream>>>(ind, ks, Kmm);
  k_cauchy_nm<<<(N_PTS * M_IND + 255) / 256, b256, 0, stream>>>(xpos, ind, ks, Knm);
  k_precT<<<(A_GP * N_PTS + 255) / 256, b256, 0, stream>>>(x_var, precT);

  // sigma_l[a] = Kmm + Knm^T diag(prec[:,a]) Knm  (batched WMMA weighted SYRK)
  wmma_gemm_tn<true, true><<<dim3(M_IND / 32, M_IND / 32, A_GP), b32, 0, stream>>>(
      Knm, Knm, precT, Kmm, Sig, N_PTS, M_IND, M_IND, 0LL, 0LL,
      (long long)N_PTS, (long long)MM);

  // Kmm inversion via Cholesky
  hipMemcpyAsync(KmmChol, Kmm, MM * sizeof(float), hipMemcpyDeviceToDevice,
                 stream);
  k_chol<<<1, b1024, 0, stream>>>(KmmChol, logdets, 0);
  k_chol<<<A_GP, b1024, 0, stream>>>(Sig, logdets, 1);
  k_trinv<<<1, b1024, 0, stream>>>(KmmChol, KmmLinv);
  k_trinv<<<A_GP, b1024, 0, stream>>>(Sig, SigLinv);

  // A^-1 = Linv^T Linv   (WMMA SYRK)
  wmma_gemm_tn<false, false><<<dim3(M_IND / 32, M_IND / 32, 1), b32, 0, stream>>>(
      KmmLinv, KmmLinv, nullptr, nullptr, KmmInv, M_IND, M_IND, M_IND, 0LL, 0LL,
      0LL, 0LL);
  wmma_gemm_tn<false, false><<<dim3(M_IND / 32, M_IND / 32, A_GP), b32, 0, stream>>>(
      SigLinv, SigLinv, nullptr, nullptr, SigInv, M_IND, M_IND, M_IND,
      (long long)MM, (long long)MM, 0LL, (long long)MM);

  k_r<<<dim3(M_IND / 256, A_GP), b256, 0, stream>>>(Knm, x_loc, precT, rv);
  k_matvec<<<dim3(M_IND / 256, A_GP), b256, 0, stream>>>(SigInv, rv, sir,
                                                         (long long)MM);
  k_gppm<<<(N_PTS * A_GP + 255) / 256, b256, 0, stream>>>(Knm, sir, gppm);

  // diag1[:,a] = rowdot(Knm @ SigInv[a], Knm)
  for (int a = 0; a < A_GP; ++a) {
    wmma_gemm_nn<<<dim3(M_IND / 32, N_PTS / 32), b32, 0, stream>>>(
        Knm, SigInv + (size_t)a * MM, Q, M_IND, M_IND);
    k_rowdot<<<N_PTS / 256, b256, 0, stream>>>(Q, Knm, diag1, M_IND, A_GP, a);
  }
  // diag2 = rowdot(Knm @ KmmInv, Knm)
  wmma_gemm_nn<<<dim3(M_IND / 32, N_PTS / 32), b32, 0, stream>>>(Knm, KmmInv, Q,
                                                                 M_IND, M_IND);
  k_rowdot<<<N_PTS / 256, b256, 0, stream>>>(Q, Knm, diag2, M_IND, 1, 0);

  // KL pieces: tr(SigInv Kmm), si_r^T Kmm si_r
  k_matvec<<<dim3(M_IND / 256, A_GP), b256, 0, stream>>>(Kmm, sir, KmmSir, 0LL);
  k_bdot<<<dim3(64, A_GP), b256, 0, stream>>>(SigInv, Kmm, acc, (int)MM,
                                              (long long)MM, 0LL, 6);
  k_bdot<<<dim3(1, A_GP), b256, 0, stream>>>(sir, KmmSir, acc, M_IND,
                                             (long long)M_IND, (long long)M_IND,
                                             14);

  k_assemble<<<(N_PTS * D_FEAT) / 256, b256, 0, stream>>>(x_loc, x_var, gppm,
                                                          diag1, diag2, out, acc);
  k_finalize<<<1, 1, 0, stream>>>(acc, logdets, out);
}